// AttentionHead_82892868812838
// MI455X (gfx1250) — compile-verified
//
#include <hip/hip_runtime.h>
#include <math.h>

// Problem constants (from reference): B=8, T=2048, D=1024, HS=64
#define B_  8
#define T_  2048
#define D_  1024
#define HS_ 64

typedef __attribute__((ext_vector_type(2))) float v2f;
typedef __attribute__((ext_vector_type(8))) float v8f;

// D = A(16x4,f32) * B(4x16,f32) + C(16x16,f32) — full-precision CDNA5 WMMA.
__device__ __forceinline__ v8f wmma4(v2f a, v2f b, v8f c) {
  return __builtin_amdgcn_wmma_f32_16x16x4_f32(
      /*neg_a=*/false, a, /*neg_b=*/false, b,
      /*c_mod=*/(short)0, c, /*reuse_a=*/false, /*reuse_b=*/false);
}

// -------------------------------------------------------------------------
// Kernel 1: q/k/v = x @ {Wq,Wk,Wv}.  One wave per 16-row tile of x.
// 12 accumulators (3 weights x 4 column tiles of 16), K-loop over D in
// steps of 4 (the f32 WMMA K).
// -------------------------------------------------------------------------
__global__ __launch_bounds__(128) void qkv_proj_kernel(
    const float* __restrict__ x,  const float* __restrict__ Wq,
    const float* __restrict__ Wk, const float* __restrict__ Wv,
    float* __restrict__ qo, float* __restrict__ ko, float* __restrict__ vo)
{
  const int wave = threadIdx.x >> 5;
  const int lane = threadIdx.x & 31;
  const int hh   = lane >> 4;      // which 16-lane half
  const int l16  = lane & 15;
  const int row0 = (blockIdx.x * 4 + wave) * 16;   // token row base, < B*T

  const float* Ws[3] = {Wq, Wk, Wv};
  float*       Os[3] = {qo, ko, vo};

  const v8f vzero = {0.f,0.f,0.f,0.f,0.f,0.f,0.f,0.f};
  v8f acc[3][4];
#pragma unroll
  for (int w = 0; w < 3; ++w)
#pragma unroll
    for (int nt = 0; nt < 4; ++nt) acc[w][nt] = vzero;

  // A-fragment base: A[M=l16][K = kk + 2*hh + v]
  const float* ap_base = x + (size_t)(row0 + l16) * D_ + 2 * hh;

  for (int kk = 0; kk < D_; kk += 4) {
    v2f a;
    a[0] = ap_base[kk];
    a[1] = ap_base[kk + 1];
#pragma unroll
    for (int w = 0; w < 3; ++w) {
      // B-fragment: B[K = kk + 2*hh + v][N = nt*16 + l16]
      const float* Wp = Ws[w] + (size_t)(kk + 2 * hh) * HS_ + l16;
#pragma unroll
      for (int nt = 0; nt < 4; ++nt) {
        v2f b;
        b[0] = Wp[nt * 16];
        b[1] = Wp[HS_ + nt * 16];
        acc[w][nt] = wmma4(a, b, acc[w][nt]);
      }
    }
  }

  // C/D layout: element j of a lane is row (j + 8*hh), column l16.
#pragma unroll
  for (int w = 0; w < 3; ++w)
#pragma unroll
    for (int nt = 0; nt < 4; ++nt)
#pragma unroll
      for (int j = 0; j < 8; ++j)
        Os[w][(size_t)(row0 + j + 8 * hh) * HS_ + nt * 16 + l16] = acc[w][nt][j];
}

// -------------------------------------------------------------------------
// Kernel 2: causal flash attention over the q/k/v workspace.
// One wave per (batch, 16-row query tile).  Online softmax with
// cross-lane shuffles; P transposed C->A layout through a per-wave LDS
// tile (no block barrier: trip counts diverge between waves).
// -------------------------------------------------------------------------
__global__ __launch_bounds__(128) void flash_attn_kernel(
    const float* __restrict__ q, const float* __restrict__ k,
    const float* __restrict__ vmat, float* __restrict__ out)
{
  __shared__ float pt[4][16][17];   // per-wave P-tile, padded vs bank conflicts

  const int wave = threadIdx.x >> 5;
  const int lane = threadIdx.x & 31;
  const int hh   = lane >> 4;
  const int l16  = lane & 15;

  const int task = blockIdx.x * 4 + wave;   // 0 .. B*(T/16)-1
  const int bidx = task >> 7;               // T/16 == 128 tiles per batch
  const int it   = task & 127;              // query tile index
  const size_t base = (size_t)bidx * T_ * HS_;
  const int qr0 = it * 16;

  // Cache the whole 16x64 q tile in registers, A-layout, 16 K-steps of 4.
  v2f aq[16];
  const float* qp = q + base + (size_t)(qr0 + l16) * HS_ + 2 * hh;
#pragma unroll
  for (int c = 0; c < 16; ++c) { aq[c][0] = qp[4 * c]; aq[c][1] = qp[4 * c + 1]; }

  const v8f vzero = {0.f,0.f,0.f,0.f,0.f,0.f,0.f,0.f};
  v8f o[4];
#pragma unroll
  for (int nt = 0; nt < 4; ++nt) o[nt] = vzero;

  float mrow[8], lrow[8];           // stats for rows (j + 8*hh)
#pragma unroll
  for (int j = 0; j < 8; ++j) { mrow[j] = -INFINITY; lrow[j] = 0.f; }

  const float scale = 0.02209708691207961f;   // 1/sqrt(2048) (block_size!)

  for (int jt = 0; jt <= it; ++jt) {
    // ---- S = q @ k^T for this 16x16 tile (B[K][N] = k[jt*16+N][K]) ----
    v8f s = vzero;
    const float* kp = k + base + (size_t)(jt * 16 + l16) * HS_ + 2 * hh;
#pragma unroll
    for (int c = 0; c < 16; ++c) {
      v2f bk; bk[0] = kp[4 * c]; bk[1] = kp[4 * c + 1];
      s = wmma4(aq[c], bk, s);
    }

    // ---- scale, causal mask, online softmax (rows live across a
    //      16-lane half -> xor 1/2/4/8 reduces a row) ----
    float corr[8];
#pragma unroll
    for (int j = 0; j < 8; ++j) {
      float sv = s[j] * scale;
      if (jt == it && l16 > j + 8 * hh) sv = -INFINITY;   // col > row
      float rm = sv;
      rm = fmaxf(rm, __shfl_xor(rm, 1, 32));
      rm = fmaxf(rm, __shfl_xor(rm, 2, 32));
      rm = fmaxf(rm, __shfl_xor(rm, 4, 32));
      rm = fmaxf(rm, __shfl_xor(rm, 8, 32));
      float mnew = fmaxf(mrow[j], rm);
      corr[j] = __expf(mrow[j] - mnew);
      float p = __expf(sv - mnew);
      mrow[j] = mnew;
      s[j] = p;
      float rs = p;
      rs += __shfl_xor(rs, 1, 32);
      rs += __shfl_xor(rs, 2, 32);
      rs += __shfl_xor(rs, 4, 32);
      rs += __shfl_xor(rs, 8, 32);
      lrow[j] = lrow[j] * corr[j] + rs;
    }

    // rescale running output
#pragma unroll
    for (int nt = 0; nt < 4; ++nt)
#pragma unroll
      for (int j = 0; j < 8; ++j) o[nt][j] *= corr[j];

    // ---- transpose P (C-layout -> A-layout) via per-wave LDS tile ----
#pragma unroll
    for (int j = 0; j < 8; ++j) pt[wave][j + 8 * hh][l16] = s[j];
    asm volatile("s_wait_dscnt 0" ::: "memory");   // same-wave DS RAW fence

    // ---- O += P @ V : 4 K-steps of 4, 4 N-tiles of 16 ----
#pragma unroll
    for (int c = 0; c < 4; ++c) {
      v2f ap;
      ap[0] = pt[wave][l16][4 * c + 2 * hh];
      ap[1] = pt[wave][l16][4 * c + 2 * hh + 1];
      const float* vrow = vmat + base + (size_t)(jt * 16 + 4 * c + 2 * hh) * HS_;
#pragma unroll
      for (int nt = 0; nt < 4; ++nt) {
        v2f bv;
        bv[0] = vrow[nt * 16 + l16];        // B[K  ][N]
        bv[1] = vrow[HS_ + nt * 16 + l16];  // B[K+1][N]
        o[nt] = wmma4(ap, bv, o[nt]);
      }
    }
  }

  // ---- normalize and store ----
#pragma unroll
  for (int j = 0; j < 8; ++j) lrow[j] = 1.0f / lrow[j];
#pragma unroll
  for (int nt = 0; nt < 4; ++nt)
#pragma unroll
    for (int j = 0; j < 8; ++j)
      out[base + (size_t)(qr0 + j + 8 * hh) * HS_ + nt * 16 + l16] =
          o[nt][j] * lrow[j];
}

// -------------------------------------------------------------------------
extern "C" void kernel_launch(void* const* d_in, const int* in_sizes, int n_in,
                              void* d_out, int out_size, void* d_ws, size_t ws_size,
                              hipStream_t stream) {
  const float* x  = (const float*)d_in[0];
  const float* Wq = (const float*)d_in[1];
  const float* Wk = (const float*)d_in[2];
  const float* Wv = (const float*)d_in[3];

  const size_t nqkv = (size_t)B_ * T_ * HS_;   // 1,048,576 floats = 4 MB each
  float* qw = (float*)d_ws;
  float* kw = qw + nqkv;
  float* vw = kw + nqkv;

  // 1024 M-tiles of 16 rows, 4 waves (128 threads) per block
  qkv_proj_kernel<<<(B_ * T_ / 16) / 4, 128, 0, stream>>>(x, Wq, Wk, Wv, qw, kw, vw);
  // 1024 (batch, q-tile) tasks, 4 waves per block
  flash_attn_kernel<<<(B_ * (T_ / 16)) / 4, 128, 0, stream>>>(qw, kw, vw, (float*)d_out);
}